// GConvSE3Partial_18743237279828
// MI455X (gfx1250) — compile-verified
//
#include <hip/hip_runtime.h>
#include <hip/hip_bf16.h>

// Problem constants (match reference)
constexpr int CB   = 16;
constexpr int CN   = 128;
constexpr int CE   = 127;
constexpr int CM   = 16;
constexpr int CBN  = CB * CN;        // 2048 nodes
constexpr int CBNE = CBN * CE;       // 260096 edges

typedef __attribute__((ext_vector_type(2))) float v2f;
typedef __attribute__((ext_vector_type(8))) float v8f;

// ---------------------------------------------------------------------------
// Stage 1: per-(node,path) MLP layers 1-2 with LayerNorm + ReLU (scalar VALU).
// One thread handles one (path, node). Output H[path][node][32].
// ---------------------------------------------------------------------------
__global__ __launch_bounds__(256)
void mlp12_kernel(const float* __restrict__ r,
                  const float* __restrict__ w1, const float* __restrict__ b1,
                  const float* __restrict__ g1, const float* __restrict__ be1,
                  const float* __restrict__ w2, const float* __restrict__ b2,
                  const float* __restrict__ g2, const float* __restrict__ be2,
                  float* __restrict__ H) {
    int tid = blockIdx.x * 256 + threadIdx.x;
    if (tid >= 4 * CBN) return;
    int p = tid >> 11;          // path 0..3
    int n = tid & (CBN - 1);    // node 0..2047

    float feat = r[(size_t)n * CE];   // first edge of node

    const float* W1 = w1 + p * 32;    // [32,1]
    const float* B1 = b1 + p * 32;
    const float* G1 = g1 + p * 32;
    const float* E1 = be1 + p * 32;
    const float* W2 = w2 + p * 1024;  // [32,32]
    const float* B2 = b2 + p * 32;
    const float* G2 = g2 + p * 32;
    const float* E2 = be2 + p * 32;

    float h[32];
    float mu = 0.f;
    #pragma unroll
    for (int c = 0; c < 32; ++c) { h[c] = fmaf(feat, W1[c], B1[c]); mu += h[c]; }
    mu *= (1.f / 32.f);
    float var = 0.f;
    #pragma unroll
    for (int c = 0; c < 32; ++c) { float d = h[c] - mu; var += d * d; }
    float inv = rsqrtf(var * (1.f / 32.f) + 1e-5f);
    #pragma unroll
    for (int c = 0; c < 32; ++c)
        h[c] = fmaxf(fmaf((h[c] - mu) * inv, G1[c], E1[c]), 0.f);

    float h2[32];
    mu = 0.f;
    #pragma unroll
    for (int c = 0; c < 32; ++c) {
        float a = B2[c];
        #pragma unroll
        for (int d = 0; d < 32; ++d) a = fmaf(W2[c * 32 + d], h[d], a);
        h2[c] = a; mu += a;
    }
    mu *= (1.f / 32.f);
    var = 0.f;
    #pragma unroll
    for (int c = 0; c < 32; ++c) { float d = h2[c] - mu; var += d * d; }
    inv = rsqrtf(var * (1.f / 32.f) + 1e-5f);

    float* out = H + ((size_t)p * CBN + n) * 32;
    #pragma unroll
    for (int c = 0; c < 32; ++c)
        out[c] = fmaxf(fmaf((h2[c] - mu) * inv, G2[c], E2[c]), 0.f);
}

// ---------------------------------------------------------------------------
// Stage 2: layer-3 GEMMs via V_WMMA_F32_16X16X4_F32 (exact fp32).
//   R_p[2048, outch] = H_p[2048,32] @ w3_p^T[32,outch] + b3_p
// One wave per (node-tile 16, out-tile 16); K=32 accumulated in 8 WMMA steps.
// A 16x4 f32 layout: lanes 0-15 K={0,1}, lanes 16-31 K={2,3} (per VGPR pair).
// B 4x16 f32 layout mirrors the K split; N = lane&15.
// D 16x16 f32 layout: VGPR v -> row M = v + 8*(lane>=16), col N = lane&15.
// ---------------------------------------------------------------------------
__global__ __launch_bounds__(256)
void wmma_l3_kernel(const float* __restrict__ Hbase,
                    const float* __restrict__ w300, const float* __restrict__ b300,
                    const float* __restrict__ w301, const float* __restrict__ b301,
                    const float* __restrict__ w310, const float* __restrict__ b310,
                    const float* __restrict__ w311, const float* __restrict__ b311,
                    float* __restrict__ R00, float* __restrict__ R01,
                    float* __restrict__ R10, float* __restrict__ R11) {
    int wave  = threadIdx.x >> 5;
    int lane  = threadIdx.x & 31;
    int t     = blockIdx.y * 8 + wave;   // global out-tile 0..95
    int ntile = blockIdx.x;              // node-tile 0..127

    const float* w3; const float* b3; float* R; int outch, ot, path;
    if (t < 16)      { path = 0; ot = t;      w3 = w300; b3 = b300; R = R00; outch = 256; }
    else if (t < 32) { path = 1; ot = t - 16; w3 = w301; b3 = b301; R = R01; outch = 256; }
    else if (t < 48) { path = 2; ot = t - 32; w3 = w310; b3 = b310; R = R10; outch = 256; }
    else             { path = 3; ot = t - 48; w3 = w311; b3 = b311; R = R11; outch = 768; }

    int lh  = lane >> 4;    // lane half: 0 or 1
    int l15 = lane & 15;

    const float* Hrow = Hbase + ((size_t)path * CBN + (size_t)ntile * 16 + l15) * 32;
    const float* Wrow = w3 + ((size_t)ot * 16 + l15) * 32;   // w3 row = out channel

    v8f acc = {};
    #pragma unroll
    for (int ks = 0; ks < 8; ++ks) {
        v2f a = *(const v2f*)(Hrow + ks * 4 + 2 * lh);   // A: rows=nodes, K slice
        v2f b = *(const v2f*)(Wrow + ks * 4 + 2 * lh);   // B: K x out-channels
        acc = __builtin_amdgcn_wmma_f32_16x16x4_f32(
            /*neg_a=*/false, a, /*neg_b=*/false, b,
            /*c_mod=*/(short)0, acc, /*reuse_a=*/false, /*reuse_b=*/false);
    }

    float bias = b3[ot * 16 + l15];
    #pragma unroll
    for (int v = 0; v < 8; ++v) {
        int node = ntile * 16 + v + 8 * lh;
        R[(size_t)node * outch + ot * 16 + l15] = acc[v] + bias;
    }
}

// ---------------------------------------------------------------------------
// Stage 3: per-node kernel contraction -> msg0[16], msg1[48].
//   msg0[i] = c00 * sum_j R00[i,j] s0[j] + sum_j R10[i,j] t1[j],
//             t1[j] = sum_k b10[k] s1[j,k]
//   msg1[i,o] = b01[o] * sum_j R01[i,j] s0[j]
//             + sum_{f,k} b11[o,k,f] * P[i][f][k],
//             P[i][f][k] = sum_j R11[i, j, f] s1[j,k]
// One thread per node.
// ---------------------------------------------------------------------------
__global__ __launch_bounds__(256)
void msg_kernel(const float* __restrict__ src0, const float* __restrict__ src1,
                const float* __restrict__ bas00, const float* __restrict__ bas01,
                const float* __restrict__ bas10, const float* __restrict__ bas11,
                const float* __restrict__ R00, const float* __restrict__ R01,
                const float* __restrict__ R10, const float* __restrict__ R11,
                float* __restrict__ msg0, float* __restrict__ msg1) {
    int n = blockIdx.x * 256 + threadIdx.x;
    if (n >= CBN) return;

    const float* S0 = src0 + (size_t)n * CE * 16;   // first edge of node
    const float* S1 = src1 + (size_t)n * CE * 48;

    float s0[16], s1[48], t1[16];
    #pragma unroll
    for (int j = 0; j < 16; ++j) s0[j] = S0[j];
    #pragma unroll
    for (int j = 0; j < 48; ++j) s1[j] = S1[j];

    float c00 = bas00[n];
    float bo1[3], bk10[3];
    #pragma unroll
    for (int k = 0; k < 3; ++k) { bo1[k] = bas01[n * 3 + k]; bk10[k] = bas10[n * 3 + k]; }
    const float* B11 = bas11 + (size_t)n * 27;      // [o][k][f]

    #pragma unroll
    for (int j = 0; j < 16; ++j)
        t1[j] = bk10[0] * s1[j * 3] + bk10[1] * s1[j * 3 + 1] + bk10[2] * s1[j * 3 + 2];

    const float* r00 = R00 + (size_t)n * 256;
    const float* r01 = R01 + (size_t)n * 256;
    const float* r10 = R10 + (size_t)n * 256;
    const float* r11 = R11 + (size_t)n * 768;       // [i][j][f], f fastest

    for (int i = 0; i < 16; ++i) {
        float a0 = 0.f, a10 = 0.f, q = 0.f;
        #pragma unroll
        for (int j = 0; j < 16; ++j) {
            a0  = fmaf(r00[i * 16 + j], s0[j], a0);
            a10 = fmaf(r10[i * 16 + j], t1[j], a10);
            q   = fmaf(r01[i * 16 + j], s0[j], q);
        }
        msg0[(size_t)n * 16 + i] = c00 * a0 + a10;

        float P[9];
        #pragma unroll
        for (int x = 0; x < 9; ++x) P[x] = 0.f;
        #pragma unroll
        for (int j = 0; j < 16; ++j) {
            #pragma unroll
            for (int f = 0; f < 3; ++f) {
                float rv = r11[i * 48 + j * 3 + f];
                P[f * 3 + 0] = fmaf(rv, s1[j * 3 + 0], P[f * 3 + 0]);
                P[f * 3 + 1] = fmaf(rv, s1[j * 3 + 1], P[f * 3 + 1]);
                P[f * 3 + 2] = fmaf(rv, s1[j * 3 + 2], P[f * 3 + 2]);
            }
        }
        #pragma unroll
        for (int o = 0; o < 3; ++o) {
            float acc = bo1[o] * q;
            #pragma unroll
            for (int k = 0; k < 3; ++k)
                #pragma unroll
                for (int f = 0; f < 3; ++f)
                    acc = fmaf(B11[o * 9 + k * 3 + f], P[f * 3 + k], acc);
            msg1[(size_t)n * 48 + i * 3 + o] = acc;
        }
    }
}

// ---------------------------------------------------------------------------
// Stage 4: broadcast per-node message to all E edges (bandwidth-critical).
// out = [ out0: BNE*16 floats | out1: BNE*48 floats ], written as float4.
// msg tables (0.5 MB) stay L2-resident; stores are fully coalesced 16B.
// ---------------------------------------------------------------------------
__global__ __launch_bounds__(256)
void bcast_kernel(const float4* __restrict__ msg0v, const float4* __restrict__ msg1v,
                  float4* __restrict__ out) {
    size_t idx = (size_t)blockIdx.x * 256 + threadIdx.x;
    const size_t n0 = (size_t)CBNE * 4;                 // float4 count of out0
    const size_t total = (size_t)CBNE * 16;             // total float4
    if (idx >= total) return;
    if (idx < n0) {
        unsigned e = (unsigned)(idx >> 2);
        unsigned q = (unsigned)(idx & 3);
        unsigned node = e / CE;
        out[idx] = msg0v[(size_t)node * 4 + q];
    } else {
        size_t j = idx - n0;
        unsigned e = (unsigned)(j / 12);
        unsigned q = (unsigned)(j % 12);
        unsigned node = e / CE;
        out[idx] = msg1v[(size_t)node * 12 + q];
    }
}

// ---------------------------------------------------------------------------
extern "C" void kernel_launch(void* const* d_in, const int* in_sizes, int n_in,
                              void* d_out, int out_size, void* d_ws, size_t ws_size,
                              hipStream_t stream) {
    const float* r    = (const float*)d_in[0];
    const float* src0 = (const float*)d_in[1];
    const float* src1 = (const float*)d_in[2];
    const float* bas00 = (const float*)d_in[3];
    const float* bas01 = (const float*)d_in[4];
    const float* bas10 = (const float*)d_in[5];
    const float* bas11 = (const float*)d_in[6];
    const float* w1  = (const float*)d_in[7];
    const float* b1  = (const float*)d_in[8];
    const float* g1  = (const float*)d_in[9];
    const float* be1 = (const float*)d_in[10];
    const float* w2  = (const float*)d_in[11];
    const float* b2  = (const float*)d_in[12];
    const float* g2  = (const float*)d_in[13];
    const float* be2 = (const float*)d_in[14];
    const float* w300 = (const float*)d_in[15];
    const float* b300 = (const float*)d_in[16];
    const float* w301 = (const float*)d_in[17];
    const float* b301 = (const float*)d_in[18];
    const float* w310 = (const float*)d_in[19];
    const float* b310 = (const float*)d_in[20];
    const float* w311 = (const float*)d_in[21];
    const float* b311 = (const float*)d_in[22];

    float* ws = (float*)d_ws;                 // all offsets 16B-aligned
    float* H    = ws;                         // 4*2048*32      = 262144
    float* R00  = ws + 262144;                // 2048*256       = 524288
    float* R01  = ws + 786432;                // 2048*256
    float* R10  = ws + 1310720;               // 2048*256
    float* R11  = ws + 1835008;               // 2048*768       = 1572864
    float* m0   = ws + 3407872;               // 2048*16
    float* m1   = ws + 3440640;               // 2048*48  (end = 3538944 floats)

    // Stage 1: MLP layers 1-2 (8192 threads)
    mlp12_kernel<<<dim3(32), 256, 0, stream>>>(r, w1, b1, g1, be1,
                                               w2, b2, g2, be2, H);

    // Stage 2: layer-3 GEMMs via WMMA f32 16x16x4
    // grid: 128 node-tiles x 12 (8 out-tiles per block, 96 total)
    wmma_l3_kernel<<<dim3(128, 12), 256, 0, stream>>>(
        H, w300, b300, w301, b301, w310, b310, w311, b311,
        R00, R01, R10, R11);

    // Stage 3: per-node contraction (2048 threads)
    msg_kernel<<<dim3(8), 256, 0, stream>>>(src0, src1, bas00, bas01, bas10, bas11,
                                            R00, R01, R10, R11, m0, m1);

    // Stage 4: broadcast to all edges (BNE*16 float4 = 66.6 MB of stores)
    const size_t total4 = (size_t)CBNE * 16;
    int blocks = (int)((total4 + 255) / 256);
    bcast_kernel<<<dim3(blocks), 256, 0, stream>>>((const float4*)m0,
                                                   (const float4*)m1,
                                                   (float4*)d_out);
}